// NBFAltNet_71133248356540
// MI455X (gfx1250) — compile-verified
//
#include <hip/hip_runtime.h>
#include <hip/hip_bf16.h>

typedef float v2f __attribute__((ext_vector_type(2)));
typedef float v8f __attribute__((ext_vector_type(8)));

#define NREP    3
#define BB      4
#define NEG     32
#define DIM     32
#define R2      100
#define LAYERS  4
#define NNODES  20000
#define BS      (NREP * BB)          /* 12 */
#define MTOT    (BS * NNODES)        /* 240000 rows of the layer GEMM */

// ---------------------------------------------------------------- utilities

__global__ void k_zero(float* __restrict__ p, long n) {
  long i = (long)blockIdx.x * blockDim.x + threadIdx.x;
  long s = (long)gridDim.x * blockDim.x;
  for (; i < n; i += s) p[i] = 0.0f;
}

__global__ void k_copy(float* __restrict__ d, const float* __restrict__ s, long n) {
  long i = (long)blockIdx.x * blockDim.x + threadIdx.x;
  long st = (long)gridDim.x * blockDim.x;
  for (; i < n; i += st) d[i] = s[i];
}

// h0 / t_neg logic for batch row b (batch is (B, NEG, 2) int32)
__device__ __forceinline__ int row_h0(const int* __restrict__ batch, int b, int* tneg_out) {
  int h00 = batch[(b * NEG + 0) * 2 + 0];
  int tneg = 1;
  for (int j = 1; j < NEG; ++j)
    if (batch[(b * NEG + j) * 2 + 0] != h00) { tneg = 0; break; }
  *tneg_out = tneg;
  return tneg ? h00 : batch[(b * NEG + 0) * 2 + 1];
}

// ---------------------------------------------------------------- init / boundary

// x[r, h0[r], :] = q[r];  pb[r, h0[r]] = 1
__global__ void k_seed(const int* __restrict__ batch, const float* __restrict__ start_query,
                       float* __restrict__ x, float* __restrict__ pb) {
  int tid = threadIdx.x;          // 384 threads
  int r = tid >> 5;
  int d = tid & 31;
  if (r >= BS) return;
  int tneg;
  int h0 = row_h0(batch, r & 3, &tneg);
  float q = start_query[(r >> 2) * DIM + d];
  x[((long)r * NNODES + h0) * DIM + d] = q;
  if (d == 0) pb[(long)r * NNODES + h0] = 1.0f;
}

// agg[r, h0[r], :] += q[r]   (agg += boundary)
__global__ void k_addboundary(const int* __restrict__ batch, const float* __restrict__ start_query,
                              float* __restrict__ agg) {
  int tid = threadIdx.x;          // 384 threads
  int r = tid >> 5;
  int d = tid & 31;
  if (r >= BS) return;
  int tneg;
  int h0 = row_h0(batch, r & 3, &tneg);
  float q = start_query[(r >> 2) * DIM + d];
  agg[((long)r * NNODES + h0) * DIM + d] += q;
}

// ---------------------------------------------------------------- rel = q @ W_rel[l] + b_rel[l]

__global__ void k_rel(const float* __restrict__ start_query, const float* __restrict__ Wl,
                      const float* __restrict__ bl, float* __restrict__ rel) {
  int idx = blockIdx.x * blockDim.x + threadIdx.x;
  if (idx >= BS * R2 * DIM) return;
  int r = idx / (R2 * DIM);
  int c = idx - r * (R2 * DIM);
  const float* q = start_query + (r >> 2) * DIM;
  float acc = bl[c];
#pragma unroll
  for (int k = 0; k < DIM; ++k) acc = fmaf(q[k], Wl[k * (R2 * DIM) + c], acc);
  rel[idx] = acc;
}

// ---------------------------------------------------------------- edge message + scatter

// one wave per edge, lane = feature dim, loop over bs rows
__global__ void k_edge(const int* __restrict__ src, const int* __restrict__ dst,
                       const int* __restrict__ etype, const float* __restrict__ edge_attr,
                       const float* __restrict__ rel, const float* __restrict__ x,
                       const float* __restrict__ pb_old,
                       float* __restrict__ agg, unsigned int* __restrict__ pb_new, int E) {
  int gid = blockIdx.x * blockDim.x + threadIdx.x;
  int e = gid >> 5;
  int lane = gid & 31;
  if (e >= E) return;
  int s = src[e], d = dst[e], t = etype[e];
  float ea = edge_attr[(long)e * DIM + lane];
#pragma unroll
  for (int r = 0; r < BS; ++r) {
    float xv = x[((long)r * NNODES + s) * DIM + lane];
    float rv = rel[(r * R2 + t) * DIM + lane];
    atomicAdd(&agg[((long)r * NNODES + d) * DIM + lane], xv * (rv + ea));
  }
  if (lane == 0) {
#pragma unroll
    for (int r = 0; r < BS; ++r) {
      unsigned int pv = __float_as_uint(pb_old[(long)r * NNODES + s]);
      atomicMax(&pb_new[(long)r * NNODES + d], pv);   // pb >= 0 -> uint order == float order
    }
  }
}

// ---------------------------------------------------------------- layer linear via WMMA f32 16x16x4
// x_new(M,32) = relu( [x | agg](M,64) @ W(64,32) + b(32) ),  M = 240000
// One wave computes a 16-row tile: two N-tiles of 16, 16 K-steps of 4 -> 32 WMMAs/tile.

__global__ __launch_bounds__(256) void k_lin(const float* __restrict__ x,
                                             const float* __restrict__ agg,
                                             const float* __restrict__ W,
                                             const float* __restrict__ bvec,
                                             float* __restrict__ xout, int Mtot) {
  __shared__ float sW[64 * 32];
  __shared__ float sB[32];
  for (int i = threadIdx.x; i < 64 * 32; i += blockDim.x) sW[i] = W[i];
  if (threadIdx.x < 32) sB[threadIdx.x] = bvec[threadIdx.x];
  __syncthreads();

  const int lane     = threadIdx.x & 31;
  const int halfSel  = lane >> 4;        // 0: lanes 0-15, 1: lanes 16-31
  const int lane16   = lane & 15;
  const int waveId   = (blockIdx.x * blockDim.x + threadIdx.x) >> 5;
  const int nWaves   = (gridDim.x * blockDim.x) >> 5;
  const int nTiles   = Mtot >> 4;        // Mtot divisible by 16

  for (int tile = waveId; tile < nTiles; tile += nWaves) {
    const int m0 = tile << 4;
    const int rowA = m0 + lane16;                 // A-matrix row for this lane
    const float* __restrict__ xrow = x   + (long)rowA * DIM;
    const float* __restrict__ arow = agg + (long)rowA * DIM;

    const float bias0 = sB[lane16];
    const float bias1 = sB[16 + lane16];
    v8f c0, c1;
#pragma unroll
    for (int i = 0; i < 8; ++i) { c0[i] = bias0; c1[i] = bias1; }

#pragma unroll
    for (int kk = 0; kk < 16; ++kk) {
      const int kbase = (kk << 2) + (halfSel << 1);   // lanes 16-31 hold K+2,K+3
      v2f a;
      if (kk < 8) { a.x = xrow[kbase];      a.y = xrow[kbase + 1]; }
      else        { a.x = arow[kbase - 32]; a.y = arow[kbase - 31]; }
      // B fragment: VGPR0 = rows K (lanes 0-15) / K+2 (lanes 16-31); VGPR1 = K+1 / K+3
      v2f b0, b1;
      b0.x = sW[kbase * 32 + lane16];
      b0.y = sW[(kbase + 1) * 32 + lane16];
      b1.x = sW[kbase * 32 + 16 + lane16];
      b1.y = sW[(kbase + 1) * 32 + 16 + lane16];
      c0 = __builtin_amdgcn_wmma_f32_16x16x4_f32(false, a, false, b0, (short)0, c0, false, false);
      c1 = __builtin_amdgcn_wmma_f32_16x16x4_f32(false, a, false, b1, (short)0, c1, false, false);
    }

    // D layout: VGPR v -> row m0+v (lanes 0-15) or m0+v+8 (lanes 16-31); col = lane16 (+16)
    const int rowBase = m0 + (halfSel << 3);
#pragma unroll
    for (int v = 0; v < 8; ++v) {
      float r0 = c0[v] > 0.0f ? c0[v] : 0.0f;
      float r1 = c1[v] > 0.0f ? c1[v] : 0.0f;
      xout[(long)(rowBase + v) * DIM + lane16]      = r0;
      xout[(long)(rowBase + v) * DIM + 16 + lane16] = r1;
    }
  }
}

// ---------------------------------------------------------------- tail: gather/mean + pb, MLP

__global__ void k_gather(const int* __restrict__ batch, const float* __restrict__ x,
                         const float* __restrict__ pb, float* __restrict__ feat,
                         float* __restrict__ out_pb) {
  int tid = threadIdx.x;          // 128 threads: (b, j)
  int b = tid >> 5;
  int j = tid & 31;
  int tneg;
  (void)row_h0(batch, b, &tneg);
  int tidx = tneg ? batch[(b * NEG + j) * 2 + 1] : batch[(b * NEG + j) * 2 + 0];

  float pbmax = 0.0f;
#pragma unroll
  for (int rep = 0; rep < NREP; ++rep)
    pbmax = fmaxf(pbmax, pb[(long)(rep * BB + b) * NNODES + tidx]);
  out_pb[b * NEG + j] = pbmax;

  for (int k = 0; k < DIM; ++k) {
    float acc = 0.0f;
#pragma unroll
    for (int rep = 0; rep < NREP; ++rep)
      acc += x[((long)(rep * BB + b) * NNODES + tidx) * DIM + k];
    feat[(b * NEG + j) * DIM + k] = acc * (1.0f / (float)NREP);
  }
}

__global__ void k_mlp(const float* __restrict__ feat,
                      const float* __restrict__ W1, const float* __restrict__ b1,
                      const float* __restrict__ W2, const float* __restrict__ b2,
                      float* __restrict__ out) {
  int row = blockIdx.x * blockDim.x + threadIdx.x;   // 128 rows
  if (row >= BB * NEG) return;
  float f[DIM], h[DIM];
#pragma unroll
  for (int k = 0; k < DIM; ++k) f[k] = feat[row * DIM + k];
  for (int d = 0; d < DIM; ++d) {
    float a = b1[d];
#pragma unroll
    for (int k = 0; k < DIM; ++k) a = fmaf(f[k], W1[k * DIM + d], a);
    h[d] = a > 0.0f ? a : 0.0f;
  }
  for (int d = 0; d < DIM; ++d) {
    float a = b2[d];
#pragma unroll
    for (int k = 0; k < DIM; ++k) a = fmaf(h[k], W2[k * DIM + d], a);
    out[row * DIM + d] = a;
  }
}

// ---------------------------------------------------------------- host orchestration

extern "C" void kernel_launch(void* const* d_in, const int* in_sizes, int n_in,
                              void* d_out, int out_size, void* d_ws, size_t ws_size,
                              hipStream_t stream) {
  const int*   batch      = (const int*)  d_in[0];
  const int*   edge_index = (const int*)  d_in[1];
  const int*   edge_type  = (const int*)  d_in[2];
  const float* edge_attr  = (const float*)d_in[3];
  const float* start_q    = (const float*)d_in[5];
  const float* W_rel      = (const float*)d_in[6];
  const float* b_rel      = (const float*)d_in[7];
  const float* W_lin      = (const float*)d_in[8];
  const float* b_lin      = (const float*)d_in[9];
  const float* W_mlp1     = (const float*)d_in[10];
  const float* b_mlp1     = (const float*)d_in[11];
  const float* W_mlp2     = (const float*)d_in[12];
  const float* b_mlp2     = (const float*)d_in[13];

  const int E = in_sizes[2];
  const int* src = edge_index;        // edge_index[0]
  const int* dst = edge_index + E;    // edge_index[1]

  // workspace layout (floats): x_a, x_b, agg, rel, pb_a, pb_b, feat  (~94 MB)
  const long xElems  = (long)BS * NNODES * DIM;   // 7,680,000
  const long pbElems = (long)BS * NNODES;         // 240,000
  float* p    = (float*)d_ws;
  float* x_a  = p; p += xElems;
  float* x_b  = p; p += xElems;
  float* agg  = p; p += xElems;
  float* rel  = p; p += (long)BS * R2 * DIM;
  float* pb_a = p; p += pbElems;
  float* pb_b = p; p += pbElems;
  float* feat = p; p += (long)BB * NEG * DIM;

  // boundary init: x = boundary (sparse), pb = indicator
  k_zero<<<2048, 256, 0, stream>>>(x_a, xElems);
  k_zero<<<256, 256, 0, stream>>>(pb_a, pbElems);
  k_seed<<<1, 384, 0, stream>>>(batch, start_q, x_a, pb_a);

  for (int l = 0; l < LAYERS; ++l) {
    k_rel<<<(BS * R2 * DIM + 255) / 256, 256, 0, stream>>>(
        start_q, W_rel + (long)l * DIM * R2 * DIM, b_rel + (long)l * R2 * DIM, rel);
    k_zero<<<2048, 256, 0, stream>>>(agg, xElems);
    k_copy<<<256, 256, 0, stream>>>(pb_b, pb_a, pbElems);
    k_edge<<<(E * 32 + 255) / 256, 256, 0, stream>>>(
        src, dst, edge_type, edge_attr, rel, x_a, pb_a, agg, (unsigned int*)pb_b, E);
    k_addboundary<<<1, 384, 0, stream>>>(batch, start_q, agg);
    k_lin<<<960, 256, 0, stream>>>(
        x_a, agg, W_lin + (long)l * 64 * DIM, b_lin + (long)l * DIM, x_b, MTOT);
    float* t;
    t = x_a;  x_a  = x_b;  x_b  = t;
    t = pb_a; pb_a = pb_b; pb_b = t;
  }

  float* out = (float*)d_out;                 // features (4096 floats) then pb (128)
  k_gather<<<1, 128, 0, stream>>>(batch, x_a, pb_a, feat, out + BB * NEG * DIM);
  k_mlp<<<1, 128, 0, stream>>>(feat, W_mlp1, b_mlp1, W_mlp2, b_mlp2, out);
}